// SymmetricContraction_34342558498837
// MI455X (gfx1250) — compile-verified
//
#include <hip/hip_runtime.h>

typedef __attribute__((ext_vector_type(2))) float v2f;
typedef __attribute__((ext_vector_type(8))) float v8f;

#define MROWS 816   // 810 useful rows (729 oij + 81 oi) padded to multiple of 16
#define KDIM  176   // 173 useful K (162 U3 + 8 U2 + 3 U1) padded to multiple of 4
#define NCOL  64    // channels k per sample = GEMM columns per workgroup
#define NSPEC 10
#define T2S   82    // t2 column stride: 81 slots + 1 dump slot
#define XS    10    // xs column stride: 9 components + constant 1.0f

// ---------------------------------------------------------------------------
// Kernel 1: build the static contraction matrix M[816][176] in workspace.
//   rows  <729 : r = o*81+i*9+j ; K<162 -> U3[o,i,j,l,a] (K=l*18+a)
//                                 162<=K<170 -> U2[o,i,j,K-162]
//   729..809   : r-729 = o*9+i  ; 170<=K<173 -> U1[o,i,K-170]
//   everything else zero (padding).
// ---------------------------------------------------------------------------
__global__ void build_M_kernel(const float* __restrict__ U1,
                               const float* __restrict__ U2,
                               const float* __restrict__ U3,
                               float* __restrict__ M) {
  int idx = blockIdx.x * blockDim.x + threadIdx.x;
  if (idx >= MROWS * KDIM) return;
  int r = idx / KDIM, kk = idx % KDIM;
  float v = 0.0f;
  if (r < 729) {
    int o = r / 81, i = (r / 9) % 9, j = r % 9;
    if (kk < 162) {
      int l = kk / 18, a = kk % 18;
      v = U3[(((o * 9 + i) * 9 + j) * 9 + l) * 18 + a];
    } else if (kk < 170) {
      v = U2[((o * 9 + i) * 9 + j) * 8 + (kk - 162)];
    }
  } else if (r < 810) {
    int oi = r - 729;
    if (kk >= 170 && kk < 173) v = U1[oi * 3 + (kk - 170)];
  }
  M[idx] = v;
}

// ---------------------------------------------------------------------------
// Kernel 2: one workgroup per sample n (species e uniform per block).
//   Phase A: wp = w[e]-projected weights (29x64) in LDS; scatter LUT built
//   Phase B: P[c][K] per-column GEMM operand in LDS (B-frag friendly layout)
//   Phase C: C = M x P via v_wmma_f32_16x16x4_f32, branchless LUT scatter
//            into t2[c][o,i] via ds_add_f32
//   Phase D: out[o] = sum_i t2[o,i] * x_i
// ---------------------------------------------------------------------------
__global__ void __launch_bounds__(256, 1)
sym_contract_kernel(const float* __restrict__ x,
                    const int* __restrict__ counts,
                    const float* __restrict__ w,
                    const float* __restrict__ proj,
                    const float* __restrict__ M,
                    float* __restrict__ out) {
  __shared__ __align__(16) float xs2[NCOL * XS];       //  2.5 KB (x + 1.0f pad)
  __shared__ __align__(16) float P[NCOL * KDIM];       // 44 KB, layout [col][K]
  __shared__ __align__(16) float scratch[NCOL * T2S];  // 20.5 KB: wp then t2
  __shared__ unsigned tab[MROWS];                      //  3.2 KB scatter LUT
  float* wp = scratch;  // 29*64 floats, dead after P is built
  float* t2 = scratch;  // 64*82 accumulators, zeroed after P is built

  const int n   = blockIdx.x;
  const int tid = threadIdx.x;

  // --- species of this sample (uniform scalar loop over 10 counts) ---------
  int e = 0;
  {
    int acc = 0;
    for (int s = 0; s < NSPEC; ++s) {
      acc += counts[s];
      if (n >= acc) e = s + 1;
    }
  }

  // --- load x tile (64 channels x 9 comps), pad each column with 1.0f ------
  for (int idx = tid; idx < NCOL * 9; idx += 256) {
    int c = idx / 9, l = idx % 9;
    xs2[c * XS + l] = x[n * NCOL * 9 + idx];
  }
  for (int c = tid; c < NCOL; c += 256) xs2[c * XS + 9] = 1.0f;

  // --- scatter LUT: tab[r] = oi | (jj << 8) --------------------------------
  //   r<729: oi = o*9+i (0..80), jj = j (0..8)  -> factor x_j
  //   729..809: oi = r-729, jj = 9              -> factor 1.0 (C1 rows)
  //   >=810 (pad, values are exactly 0): oi = 81 (dump), jj = 9
  for (int r = tid; r < MROWS; r += 256) {
    unsigned oi, jj;
    if (r < 729)      { oi = (unsigned)((r / 81) * 9 + (r / 9) % 9); jj = (unsigned)(r % 9); }
    else if (r < 810) { oi = (unsigned)(r - 729); jj = 9u; }
    else              { oi = 81u; jj = 9u; }
    tab[r] = oi | (jj << 8);
  }

  // --- wp[b][c] = sum_a w[e,a,c] * proj[a,b]  (29x64) ----------------------
  for (int idx = tid; idx < 29 * NCOL; idx += 256) {
    int b = idx / NCOL, c = idx % NCOL;
    float s = 0.0f;
    for (int a = 0; a < 29; ++a)
      s += w[(e * 29 + a) * NCOL + c] * proj[a * 29 + b];
    wp[b * NCOL + c] = s;
  }
  __syncthreads();

  // --- build P[c][K]: columns of the GEMM B operand ------------------------
  // w1 = wp rows 0..2, w2 = wp rows 3..10, w3 = wp rows 11..28
  for (int idx = tid; idx < NCOL * KDIM; idx += 256) {
    int c = idx / KDIM, kk = idx % KDIM;
    float v = 0.0f;
    if (kk < 162) {
      int l = kk / 18, a = kk % 18;
      v = xs2[c * XS + l] * wp[(11 + a) * NCOL + c];   // x_l * w3_a
    } else if (kk < 170) {
      v = wp[(3 + (kk - 162)) * NCOL + c];             // w2_a
    } else if (kk < 173) {
      v = wp[(kk - 170) * NCOL + c];                   // w1_a
    }
    P[c * KDIM + kk] = v;
  }
  __syncthreads();

  // --- retire wp, zero t2 accumulators -------------------------------------
  for (int idx = tid; idx < NCOL * T2S; idx += 256) t2[idx] = 0.0f;
  __syncthreads();

  // --- fp32 WMMA GEMM: C[816x64] = M[816x176] * P^T[176x64], fused epilogue
  // f32 16x16x4 frag mapping: A/B lane holds K = {0,1}+2*(lane>=16), row/col = lane%16
  const int wave = tid >> 5;
  const int lane = tid & 31;
  const int lm   = lane & 15;
  const int hi   = lane >> 4;

  for (int rt = wave; rt < MROWS / 16; rt += 8) {
    v8f acc0 = {}, acc1 = {}, acc2 = {}, acc3 = {};
    const float* Arow = M + (rt * 16 + lm) * KDIM + 2 * hi;  // global, L2-resident
    const float* Pb   = &P[lm * KDIM + 2 * hi];              // LDS, b64 per frag
    for (int k0 = 0; k0 < KDIM; k0 += 4) {
      v2f a  = *(const v2f*)(Arow + k0);
      v2f b0 = *(const v2f*)(Pb + 0 * 16 * KDIM + k0);
      v2f b1 = *(const v2f*)(Pb + 1 * 16 * KDIM + k0);
      v2f b2 = *(const v2f*)(Pb + 2 * 16 * KDIM + k0);
      v2f b3 = *(const v2f*)(Pb + 3 * 16 * KDIM + k0);
      acc0 = __builtin_amdgcn_wmma_f32_16x16x4_f32(false, a, false, b0, (short)0, acc0, false, false);
      acc1 = __builtin_amdgcn_wmma_f32_16x16x4_f32(false, a, false, b1, (short)0, acc1, false, false);
      acc2 = __builtin_amdgcn_wmma_f32_16x16x4_f32(false, a, false, b2, (short)0, acc2, false, false);
      acc3 = __builtin_amdgcn_wmma_f32_16x16x4_f32(false, a, false, b3, (short)0, acc3, false, false);
    }
    // Branchless epilogue. C/D tile layout: lane holds
    //   (row = rt*16 + m + 8*hi, col = ct*16 + lm) in acc element m.
    // LUT entries are shared across the 4 column tiles.
    const int rbase = rt * 16 + 8 * hi;
    unsigned te[8];
    for (int m = 0; m < 8; ++m) te[m] = tab[rbase + m];
    v8f accs[4] = {acc0, acc1, acc2, acc3};
    for (int ct = 0; ct < 4; ++ct) {
      const int c = ct * 16 + lm;
      union { v8f v; float f[8]; } u;
      u.v = accs[ct];
      float*       t2c = &t2[c * T2S];
      const float* xc  = &xs2[c * XS];
      for (int m = 0; m < 8; ++m) {
        unsigned t = te[m];
        atomicAdd(&t2c[t & 0xFFu], u.f[m] * xc[(t >> 8) & 0xFFu]);
      }
    }
  }
  __syncthreads();

  // --- final contraction: out[n,c,o] = sum_i t2[c][o,i] * x_i --------------
  for (int idx = tid; idx < NCOL * 9; idx += 256) {
    int c = idx / 9, o = idx % 9;
    float s = 0.0f;
    for (int i = 0; i < 9; ++i)
      s += t2[c * T2S + o * 9 + i] * xs2[c * XS + i];
    out[n * NCOL * 9 + idx] = s;
  }
}

// ---------------------------------------------------------------------------
extern "C" void kernel_launch(void* const* d_in, const int* in_sizes, int n_in,
                              void* d_out, int out_size, void* d_ws, size_t ws_size,
                              hipStream_t stream) {
  const float* x    = (const float*)d_in[0];
  const int*   cnts = (const int*)d_in[1];
  const float* w    = (const float*)d_in[2];
  const float* proj = (const float*)d_in[3];
  const float* U1   = (const float*)d_in[4];
  const float* U2   = (const float*)d_in[5];
  const float* U3   = (const float*)d_in[6];
  float* out = (float*)d_out;
  float* M   = (float*)d_ws;  // 816*176*4 = 574,464 bytes

  const int n_samples = in_sizes[0] / (NCOL * 9);  // 1024

  int mtotal = MROWS * KDIM;
  build_M_kernel<<<(mtotal + 255) / 256, 256, 0, stream>>>(U1, U2, U3, M);
  sym_contract_kernel<<<n_samples, 256, 0, stream>>>(x, cnts, w, proj, M, out);
}